// ConcreteSelect_45011257262583
// MI455X (gfx1250) — compile-verified
//
#include <hip/hip_runtime.h>
#include <hip/hip_bf16.h>

typedef __attribute__((ext_vector_type(2))) float v2f;
typedef __attribute__((ext_vector_type(8))) float v8f;

#define B_ROWS 8192
#define S_ROWS 256
#define D_DIM  10000
#define HALF_N 1280000u      // (S_ROWS*D_DIM)/2, threefry counter split point
#define TINY_F 1.1754943508222875e-38f

// ---------------------------------------------------------------------------
// JAX threefry2x32, key = jax.random.key(42) -> (hi=0, lo=42).
// counts = iota(2*HALF_N); x0 = counts[p], x1 = counts[p+HALF_N];
// flat element e: p = e % HALF_N, component = e / HALF_N.
// ---------------------------------------------------------------------------
__device__ __forceinline__ unsigned rotl32(unsigned x, int r) {
    return (x << r) | (x >> (32 - r));
}

__device__ __forceinline__ unsigned threefry_bits(unsigned e) {
    const unsigned ks0 = 0u, ks1 = 42u, ks2 = 0x1BD11BDAu ^ 0u ^ 42u;
    unsigned p    = (e < HALF_N) ? e : (e - HALF_N);
    unsigned comp = (e < HALF_N) ? 0u : 1u;
    unsigned x0 = p + ks0;
    unsigned x1 = (p + HALF_N) + ks1;
#define TF_ROUND(r) { x0 += x1; x1 = rotl32(x1, r); x1 ^= x0; }
    TF_ROUND(13) TF_ROUND(15) TF_ROUND(26) TF_ROUND(6)   x0 += ks1; x1 += ks2 + 1u;
    TF_ROUND(17) TF_ROUND(29) TF_ROUND(16) TF_ROUND(24)  x0 += ks2; x1 += ks0 + 2u;
    TF_ROUND(13) TF_ROUND(15) TF_ROUND(26) TF_ROUND(6)   x0 += ks0; x1 += ks1 + 3u;
    TF_ROUND(17) TF_ROUND(29) TF_ROUND(16) TF_ROUND(24)  x0 += ks1; x1 += ks2 + 4u;
    TF_ROUND(13) TF_ROUND(15) TF_ROUND(26) TF_ROUND(6)   x0 += ks2; x1 += ks0 + 5u;
#undef TF_ROUND
    return comp ? x1 : x0;
}

__device__ __forceinline__ float gumbel_at(unsigned e) {
    unsigned bits = threefry_bits(e);
    unsigned fb = (bits >> 9) | 0x3f800000u;
    float u = __uint_as_float(fb) - 1.0f;            // [0,1)
    u = u * (1.0f - TINY_F) + TINY_F;                // jax uniform(minval=tiny, maxval=1)
    u = fmaxf(TINY_F, u);
    return -logf(-logf(u));
}

// ---------------------------------------------------------------------------
// Kernel 1: per logits row s -> idx[s] = argmax(z), c[s] = (1-y)+y, y = 1/sum(exp(z-zmax))
// One block per row, z staged in LDS (40 KB << 320 KB per WGP).
// ---------------------------------------------------------------------------
__global__ __launch_bounds__(256)
void gumbel_argmax_kernel(const float* __restrict__ logits,
                          const int*  __restrict__ temp,
                          int*  __restrict__ wsIdx,
                          float* __restrict__ wsC) {
    __shared__ float zbuf[D_DIM];
    __shared__ float rv[256];
    __shared__ int   ri[256];

    const int s   = blockIdx.x;
    const int tid = threadIdx.x;
    const float invTemp = 1.0f / (float)temp[0];
    const float* lrow = logits + (size_t)s * D_DIM;

    float bestv = -3.402823466e+38f;
    int   besti = 0;
    for (int d = tid; d < D_DIM; d += 256) {
        unsigned e = (unsigned)(s * D_DIM + d);
        float z = (lrow[d] + gumbel_at(e)) * invTemp;
        zbuf[d] = z;
        if (z > bestv) { bestv = z; besti = d; }
    }
    rv[tid] = bestv; ri[tid] = besti;
    __syncthreads();
    for (int off = 128; off > 0; off >>= 1) {
        if (tid < off) {
            float ov = rv[tid + off]; int oi = ri[tid + off];
            if (ov > rv[tid] || (ov == rv[tid] && oi < ri[tid])) { rv[tid] = ov; ri[tid] = oi; }
        }
        __syncthreads();
    }
    const float zmax = rv[0];
    const int   amax = ri[0];
    __syncthreads();

    float partial = 0.0f;
    for (int d = tid; d < D_DIM; d += 256) partial += expf(zbuf[d] - zmax);
    rv[tid] = partial;
    __syncthreads();
    for (int off = 128; off > 0; off >>= 1) {
        if (tid < off) rv[tid] += rv[tid + off];
        __syncthreads();
    }
    if (tid == 0) {
        float y = 1.0f / rv[0];
        wsIdx[s] = amax;
        wsC[s]   = (1.0f - y) + y;   // matches y_hard - sg(y_soft) + y_soft at argmax
    }
}

// ---------------------------------------------------------------------------
// Kernel 2: Y = X_gathered @ diag(c) via V_WMMA_F32_16X16X4_F32.
// grid.x = 512 (16-row tiles of B), 8 waves/block, each wave owns 2 n-tiles.
// A 16x4 layout: lanes 0-15 -> K0(V0),K1(V1); lanes 16-31 -> K2,K3.
// B 4x16 diagonal block; C/D 16x16: VGPR v -> M=v (lanes 0-15) / M=v+8.
// ---------------------------------------------------------------------------
__global__ __launch_bounds__(256)
void gather_wmma_kernel(const float* __restrict__ X,
                        const int*   __restrict__ wsIdx,
                        const float* __restrict__ wsC,
                        float* __restrict__ Y) {
    __shared__ int   sIdx[S_ROWS];
    __shared__ float sC[S_ROWS];

    const int tid = threadIdx.x;
    sIdx[tid] = wsIdx[tid];
    sC[tid]   = wsC[tid];
    __syncthreads();

    const int lane = tid & 31;
    const int wave = tid >> 5;
    const int M    = lane & 15;
    const int half = lane >> 4;
    const long b0  = (long)blockIdx.x * 16;
    const float* Xrow = X + (b0 + M) * (long)D_DIM;

    #pragma unroll
    for (int t = 0; t < 2; ++t) {
        const int n  = wave * 2 + t;
        const int nb = n * 16;
        const int N  = M;                 // C/D column index for this lane
        const float cval = sC[nb + N];

        v8f acc = {};
        #pragma unroll
        for (int kk = 0; kk < 4; ++kk) {
            const int k0 = kk * 4 + 2 * half;   // K index of A VGPR0 for this lane half
            v2f a, b;
            a.x = Xrow[sIdx[nb + k0]];
            a.y = Xrow[sIdx[nb + k0 + 1]];
            b.x = (N == k0)     ? cval : 0.0f;  // diagonal block of diag(c)
            b.y = (N == k0 + 1) ? cval : 0.0f;
            acc = __builtin_amdgcn_wmma_f32_16x16x4_f32(
                false, a, false, b, (short)0, acc, false, false);
        }

        float* Yt = Y + (b0 + 8 * half) * S_ROWS + nb + N;
        #pragma unroll
        for (int v = 0; v < 8; ++v) Yt[v * S_ROWS] = acc[v];
    }
}

// ---------------------------------------------------------------------------
extern "C" void kernel_launch(void* const* d_in, const int* in_sizes, int n_in,
                              void* d_out, int out_size, void* d_ws, size_t ws_size,
                              hipStream_t stream) {
    const float* X      = (const float*)d_in[0];   // (8192, 10000)
    const float* logits = (const float*)d_in[1];   // (256, 10000)
    const int*   temp   = (const int*)d_in[2];     // scalar
    float* Y = (float*)d_out;                      // (8192, 256)

    int*   wsIdx = (int*)d_ws;
    float* wsC   = (float*)d_ws + S_ROWS;

    gumbel_argmax_kernel<<<S_ROWS, 256, 0, stream>>>(logits, temp, wsIdx, wsC);
    gather_wmma_kernel<<<B_ROWS / 16, 256, 0, stream>>>(X, wsIdx, wsC, Y);
}